// DrugTargetGNN_10788957847627
// MI455X (gfx1250) — compile-verified
//
#include <hip/hip_runtime.h>
#include <math.h>

typedef __attribute__((ext_vector_type(2))) float v2f;
typedef __attribute__((ext_vector_type(8))) float v8f;

#define G_BATCH 512

// ---------------------------------------------------------------------------
// Fused GEMM: Out[M,N] = opt_relu(X[M,K] @ W[K,N] + bias)   (all row-major)
// Block = 128 threads = 4 waves. Each wave computes a 64x16 output slab
// (4 stacked 16x16 WMMA tiles): one B fragment feeds 4 WMMAs per k-step.
// Software-pipelined: fragments for step k+4 are loaded before the WMMAs of
// step k are issued, so waits before WMMA are partial, not full drains.
// ---------------------------------------------------------------------------
__global__ __launch_bounds__(128)
void gemm_wmma_f32(const float* __restrict__ X, const float* __restrict__ W,
                   const float* __restrict__ bias, float* __restrict__ Out,
                   int M, int K, int N, int ldout, int do_relu)
{
    const int lane = threadIdx.x & 31;
    const int wav  = threadIdx.x >> 5;                 // 0..3
    const int l15  = lane & 15;
    const int hi   = lane >> 4;                        // 0 -> K{0,1}, 1 -> K{2,3}
    const int koff = 2 * hi;

    const int ncol  = blockIdx.x * 16 + l15;
    const int ncolc = (ncol < N) ? ncol : (N - 1);     // clamped (safe, unstored)
    const int mtop  = (blockIdx.y * 4 + wav) * 64;     // 64 rows per wave

    // Clamped row base pointers for the 4 sub-tiles (A fragments).
    const float* Xr[4];
#pragma unroll
    for (int s = 0; s < 4; ++s) {
        int m  = mtop + s * 16 + l15;
        int mc = (m < M) ? m : (M - 1);
        Xr[s] = X + (size_t)mc * K;
    }

    v8f acc[4] = {};

    const int K4 = K & ~3;
    const float* Wp = W + (size_t)koff * N + ncolc;    // bumped by 4N per step

    if (K4 > 0) {
        // ---- prologue: load fragments for k = 0 ----
        v2f b, a0, a1, a2, a3;
        b.x = Wp[0];
        b.y = Wp[N];
        a0.x = Xr[0][koff]; a0.y = Xr[0][koff + 1];
        a1.x = Xr[1][koff]; a1.y = Xr[1][koff + 1];
        a2.x = Xr[2][koff]; a2.y = Xr[2][koff + 1];
        a3.x = Xr[3][koff]; a3.y = Xr[3][koff + 1];

        // ---- pipelined main loop: prefetch k+4, then WMMA on k ----
        for (int k = 0; k + 4 < K4; k += 4) {
            const float* Wn = Wp + (size_t)4 * N;
            v2f bn, n0, n1, n2, n3;
            bn.x = Wn[0];
            bn.y = Wn[N];
            const int kkn = k + 4 + koff;
            n0.x = Xr[0][kkn]; n0.y = Xr[0][kkn + 1];
            n1.x = Xr[1][kkn]; n1.y = Xr[1][kkn + 1];
            n2.x = Xr[2][kkn]; n2.y = Xr[2][kkn + 1];
            n3.x = Xr[3][kkn]; n3.y = Xr[3][kkn + 1];

            acc[0] = __builtin_amdgcn_wmma_f32_16x16x4_f32(false, a0, false, b, (short)0, acc[0], false, false);
            acc[1] = __builtin_amdgcn_wmma_f32_16x16x4_f32(false, a1, false, b, (short)0, acc[1], false, false);
            acc[2] = __builtin_amdgcn_wmma_f32_16x16x4_f32(false, a2, false, b, (short)0, acc[2], false, false);
            acc[3] = __builtin_amdgcn_wmma_f32_16x16x4_f32(false, a3, false, b, (short)0, acc[3], false, false);

            b = bn; a0 = n0; a1 = n1; a2 = n2; a3 = n3;
            Wp = Wn;
        }

        // ---- epilogue: last full k-step ----
        acc[0] = __builtin_amdgcn_wmma_f32_16x16x4_f32(false, a0, false, b, (short)0, acc[0], false, false);
        acc[1] = __builtin_amdgcn_wmma_f32_16x16x4_f32(false, a1, false, b, (short)0, acc[1], false, false);
        acc[2] = __builtin_amdgcn_wmma_f32_16x16x4_f32(false, a2, false, b, (short)0, acc[2], false, false);
        acc[3] = __builtin_amdgcn_wmma_f32_16x16x4_f32(false, a3, false, b, (short)0, acc[3], false, false);
    }

    if (K4 < K) {  // K % 4 == 2 (e.g. K = 78): zero B where k >= K, clamp A idx
        const int k0 = K4 + koff, k1 = k0 + 1;
        const int k0c = (k0 < K) ? k0 : 0;
        const int k1c = (k1 < K) ? k1 : 0;
        float bx = W[(size_t)k0c * N + ncolc];
        float by = W[(size_t)k1c * N + ncolc];
        v2f b;
        b.x = (k0 < K) ? bx : 0.f;
        b.y = (k1 < K) ? by : 0.f;
        v2f a0, a1, a2, a3;
        a0.x = Xr[0][k0c]; a0.y = Xr[0][k1c];
        a1.x = Xr[1][k0c]; a1.y = Xr[1][k1c];
        a2.x = Xr[2][k0c]; a2.y = Xr[2][k1c];
        a3.x = Xr[3][k0c]; a3.y = Xr[3][k1c];
        acc[0] = __builtin_amdgcn_wmma_f32_16x16x4_f32(false, a0, false, b, (short)0, acc[0], false, false);
        acc[1] = __builtin_amdgcn_wmma_f32_16x16x4_f32(false, a1, false, b, (short)0, acc[1], false, false);
        acc[2] = __builtin_amdgcn_wmma_f32_16x16x4_f32(false, a2, false, b, (short)0, acc[2], false, false);
        acc[3] = __builtin_amdgcn_wmma_f32_16x16x4_f32(false, a3, false, b, (short)0, acc[3], false, false);
    }

    if (ncol < N) {
        const float bv = bias ? bias[ncol] : 0.f;
        if (mtop + 64 <= M) {          // fast path: whole slab in range
#pragma unroll
            for (int s = 0; s < 4; ++s) {
                const int mbase = mtop + s * 16 + hi * 8;
#pragma unroll
                for (int r = 0; r < 8; ++r) {
                    float v = acc[s][r] + bv;
                    if (do_relu) v = fmaxf(v, 0.f);
                    Out[(size_t)(mbase + r) * ldout + ncol] = v;
                }
            }
        } else {
#pragma unroll
            for (int s = 0; s < 4; ++s) {
                const int mbase = mtop + s * 16 + hi * 8;
#pragma unroll
                for (int r = 0; r < 8; ++r) {
                    const int m = mbase + r;
                    if (m < M) {
                        float v = acc[s][r] + bv;
                        if (do_relu) v = fmaxf(v, 0.f);
                        Out[(size_t)m * ldout + ncol] = v;
                    }
                }
            }
        }
    }
}

// ---------------------------------------------------------------------------
// Degree / normalization helpers
// ---------------------------------------------------------------------------
__global__ void deg_init(float* __restrict__ deg, int n)
{
    int i = blockIdx.x * blockDim.x + threadIdx.x;
    if (i < n) deg[i] = 1.0f;              // self-loop contribution
}

__global__ void deg_accum(const int* __restrict__ cols, float* __restrict__ deg, int e)
{
    int i = blockIdx.x * blockDim.x + threadIdx.x;
    if (i < e) atomicAdd(&deg[cols[i]], 1.0f);
}

__global__ void deg_rsqrt_inplace(float* __restrict__ d, int n)
{
    int i = blockIdx.x * blockDim.x + threadIdx.x;
    if (i < n) d[i] = rsqrtf(fmaxf(d[i], 1e-12f));
}

// ---------------------------------------------------------------------------
// GCN propagation: self-loop init, per-edge scatter-add, relu
// ---------------------------------------------------------------------------
__global__ void selfloop_init(const float* __restrict__ H, const float* __restrict__ dinv,
                              float* __restrict__ Out, int Nn, int F)
{
    size_t i = (size_t)blockIdx.x * blockDim.x + threadIdx.x;
    if (i >= (size_t)Nn * F) return;
    int n = (int)(i / (unsigned)F);
    float d = dinv[n];
    Out[i] = H[i] * d * d;
}

// One wave per edge; features vectorized as float2 (all F here are even).
__global__ void scatter_edges(const float* __restrict__ H, const int* __restrict__ rows,
                              const int* __restrict__ cols, const float* __restrict__ eattr,
                              const float* __restrict__ dinv, float* __restrict__ Out,
                              int E, int F)
{
    int w    = (int)((blockIdx.x * (size_t)blockDim.x + threadIdx.x) >> 5);
    int lane = threadIdx.x & 31;
    if (w >= E) return;
    int r = rows[w], c = cols[w];
    float norm = dinv[r] * eattr[w] * dinv[c];
    const float2* hr = (const float2*)(H + (size_t)r * F);
    float*        oc = Out + (size_t)c * F;
    const int F2 = F >> 1;
    for (int f = lane; f < F2; f += 32) {
        float2 hv = hr[f];
        atomicAdd(&oc[2 * f + 0], hv.x * norm);
        atomicAdd(&oc[2 * f + 1], hv.y * norm);
    }
}

__global__ void relu_inplace(float* __restrict__ x, size_t n)
{
    size_t i = (size_t)blockIdx.x * blockDim.x + threadIdx.x;
    if (i < n) x[i] = fmaxf(x[i], 0.f);
}

// ---------------------------------------------------------------------------
// Mean pool
// ---------------------------------------------------------------------------
__global__ void zero_f32(float* __restrict__ p, size_t n)
{
    size_t i = (size_t)blockIdx.x * blockDim.x + threadIdx.x;
    if (i < n) p[i] = 0.f;
}

__global__ void pool_accum(const float* __restrict__ X, const int* __restrict__ batch,
                           float* __restrict__ pooled, float* __restrict__ cnt,
                           int Nn, int F)
{
    int w    = (int)((blockIdx.x * (size_t)blockDim.x + threadIdx.x) >> 5); // wave = node
    int lane = threadIdx.x & 31;
    if (w >= Nn) return;
    int b = batch[w];
    const float* xr = X      + (size_t)w * F;
    float*       pr = pooled + (size_t)b * F;
    for (int f = lane; f < F; f += 32) atomicAdd(&pr[f], xr[f]);
    if (lane == 0) atomicAdd(&cnt[b], 1.0f);
}

__global__ void pool_div(float* __restrict__ pooled, const float* __restrict__ cnt,
                         int G, int F)
{
    int i = blockIdx.x * blockDim.x + threadIdx.x;
    if (i >= G * F) return;
    pooled[i] /= fmaxf(cnt[i / F], 1.0f);
}

// ---------------------------------------------------------------------------
// Final projection: out[g] = H[g,:] @ w + b    (wave-per-row reduction)
// ---------------------------------------------------------------------------
__global__ __launch_bounds__(32)
void final_dot(const float* __restrict__ H, const float* __restrict__ w,
               const float* __restrict__ b, float* __restrict__ out, int K)
{
    int g = blockIdx.x, lane = threadIdx.x;
    float s = 0.f;
    for (int k = lane; k < K; k += 32) s += H[(size_t)g * K + k] * w[k];
    for (int off = 16; off > 0; off >>= 1) s += __shfl_down(s, off, 32);
    if (lane == 0) out[g] = s + b[0];
}

// ---------------------------------------------------------------------------
// Host orchestration
// ---------------------------------------------------------------------------
static inline unsigned cdiv(size_t a, size_t b) { return (unsigned)((a + b - 1) / b); }

extern "C" void kernel_launch(void* const* d_in, const int* in_sizes, int n_in,
                              void* d_out, int out_size, void* d_ws, size_t ws_size,
                              hipStream_t stream)
{
    // ---- inputs (setup_inputs dict order) ----
    const float* drug_x   = (const float*)d_in[0];
    const int*   d_ei     = (const int*)  d_in[1];
    const float* d_ea     = (const float*)d_in[2];
    const int*   d_batch  = (const int*)  d_in[3];
    const float* prot_x   = (const float*)d_in[4];
    const int*   p_ei     = (const int*)  d_in[5];
    const float* p_ea     = (const float*)d_in[6];
    const int*   p_batch  = (const int*)  d_in[7];
    const float* dW1  = (const float*)d_in[8];
    const float* dW2  = (const float*)d_in[9];
    const float* dW3  = (const float*)d_in[10];
    const float* dL1w = (const float*)d_in[11];
    const float* dL1b = (const float*)d_in[12];
    const float* dL2w = (const float*)d_in[13];
    const float* dL2b = (const float*)d_in[14];
    const float* pW1  = (const float*)d_in[15];
    const float* pW2  = (const float*)d_in[16];
    const float* pW3  = (const float*)d_in[17];
    const float* pL1w = (const float*)d_in[18];
    const float* pL1b = (const float*)d_in[19];
    const float* pL2w = (const float*)d_in[20];
    const float* pL2b = (const float*)d_in[21];
    const float* fW1  = (const float*)d_in[22];
    const float* fb1  = (const float*)d_in[23];
    const float* fW2  = (const float*)d_in[24];
    const float* fb2  = (const float*)d_in[25];
    const float* fW3  = (const float*)d_in[26];
    const float* fb3  = (const float*)d_in[27];
    float* out = (float*)d_out;

    const int Nd = in_sizes[0] / 78;
    const int Ed = in_sizes[1] / 2;
    const int Np = in_sizes[4] / 20;
    const int Ep = in_sizes[5] / 2;
    const int G  = G_BATCH;

    const int* d_rows = d_ei;            // edge_index[0]
    const int* d_cols = d_ei + Ed;       // edge_index[1]
    const int* p_rows = p_ei;
    const int* p_cols = p_ei + Ep;

    // ---- workspace carve-out ----
    float* wsp = (float*)d_ws;
    size_t bigN = (size_t)Nd * 312;
    size_t bigP = (size_t)Np * 80;
    if (bigP > bigN) bigN = bigP;
    float* big0   = wsp;           wsp += bigN;      // GEMM outputs (h)
    float* big1   = wsp;           wsp += bigN;      // scatter outputs (s)
    float* dinv_d = wsp;           wsp += Nd;
    float* dinv_p = wsp;           wsp += Np;
    float* dpool  = wsp;           wsp += (size_t)G * 312;
    float* ppool  = wsp;           wsp += (size_t)G * 80;
    float* cntd   = wsp;           wsp += G;
    float* cntp   = wsp;           wsp += G;
    float* h1     = wsp;           wsp += (size_t)G * 1024;  // reused 3x
    float* fh2    = wsp;           wsp += (size_t)G * 512;
    float* fused  = wsp;           wsp += (size_t)G * 128;
    (void)ws_size; (void)n_in; (void)out_size;

    const int T = 256;

    // ---- degree -> dinv (shared by all 3 layers per graph type) ----
    deg_init<<<cdiv(Nd, T), T, 0, stream>>>(dinv_d, Nd);
    deg_accum<<<cdiv(Ed, T), T, 0, stream>>>(d_cols, dinv_d, Ed);
    deg_rsqrt_inplace<<<cdiv(Nd, T), T, 0, stream>>>(dinv_d, Nd);

    deg_init<<<cdiv(Np, T), T, 0, stream>>>(dinv_p, Np);
    deg_accum<<<cdiv(Ep, T), T, 0, stream>>>(p_cols, dinv_p, Ep);
    deg_rsqrt_inplace<<<cdiv(Np, T), T, 0, stream>>>(dinv_p, Np);

    auto gemm = [&](const float* X, const float* W, const float* b, float* O,
                    int M, int K, int N, int ld, int relu) {
        dim3 grid(cdiv(N, 16), cdiv(M, 256));
        gemm_wmma_f32<<<grid, 128, 0, stream>>>(X, W, b, O, M, K, N, ld, relu);
    };

    // ---- one GCN layer: t=x@W; s=selfloop+scatter(t); relu(s) ----
    auto gcn = [&](const float* x, const float* Wm, int Nn, int Fin, int Fout,
                   const int* rows, const int* cols, const float* ea, int E,
                   const float* dinv, float* t, float* s) {
        gemm(x, Wm, nullptr, t, Nn, Fin, Fout, Fout, 0);
        size_t tot = (size_t)Nn * Fout;
        selfloop_init<<<cdiv(tot, T), T, 0, stream>>>(t, dinv, s, Nn, Fout);
        scatter_edges<<<cdiv((size_t)E * 32, T), T, 0, stream>>>(t, rows, cols, ea, dinv, s, E, Fout);
        relu_inplace<<<cdiv(tot, T), T, 0, stream>>>(s, tot);
    };

    // ---- drug branch ----
    gcn(drug_x, dW1, Nd, 78,  78,  d_rows, d_cols, d_ea, Ed, dinv_d, big0, big1);
    gcn(big1,   dW2, Nd, 78,  156, d_rows, d_cols, d_ea, Ed, dinv_d, big0, big1);
    gcn(big1,   dW3, Nd, 156, 312, d_rows, d_cols, d_ea, Ed, dinv_d, big0, big1);

    zero_f32<<<cdiv((size_t)G * 312, T), T, 0, stream>>>(dpool, (size_t)G * 312);
    zero_f32<<<cdiv((size_t)G, T), T, 0, stream>>>(cntd, G);
    pool_accum<<<cdiv((size_t)Nd * 32, T), T, 0, stream>>>(big1, d_batch, dpool, cntd, Nd, 312);
    pool_div<<<cdiv((size_t)G * 312, T), T, 0, stream>>>(dpool, cntd, G, 312);

    gemm(dpool, dL1w, dL1b, h1, G, 312, 1024, 1024, 1);
    gemm(h1, dL2w, dL2b, fused /*cols 0..63*/, G, 1024, 64, 128, 1);

    // ---- protein branch (reuses big0/big1/h1) ----
    gcn(prot_x, pW1, Np, 20, 20, p_rows, p_cols, p_ea, Ep, dinv_p, big0, big1);
    gcn(big1,   pW2, Np, 20, 40, p_rows, p_cols, p_ea, Ep, dinv_p, big0, big1);
    gcn(big1,   pW3, Np, 40, 80, p_rows, p_cols, p_ea, Ep, dinv_p, big0, big1);

    zero_f32<<<cdiv((size_t)G * 80, T), T, 0, stream>>>(ppool, (size_t)G * 80);
    zero_f32<<<cdiv((size_t)G, T), T, 0, stream>>>(cntp, G);
    pool_accum<<<cdiv((size_t)Np * 32, T), T, 0, stream>>>(big1, p_batch, ppool, cntp, Np, 80);
    pool_div<<<cdiv((size_t)G * 80, T), T, 0, stream>>>(ppool, cntp, G, 80);

    gemm(ppool, pL1w, pL1b, h1, G, 80, 1024, 1024, 1);
    gemm(h1, pL2w, pL2b, fused + 64 /*cols 64..127*/, G, 1024, 64, 128, 1);

    // ---- fusion head ----
    gemm(fused, fW1, fb1, h1, G, 128, 1024, 1024, 1);
    gemm(h1, fW2, fb2, fh2, G, 1024, 512, 512, 1);
    final_dot<<<G, 32, 0, stream>>>(fh2, fW3, fb3, out, 512);
}